// GCNNet_73512660238714
// MI455X (gfx1250) — compile-verified
//
#include <hip/hip_runtime.h>
#include <hip/hip_bf16.h>

typedef __attribute__((ext_vector_type(2))) float v2f;
typedef __attribute__((ext_vector_type(8))) float v8f;

#define FEAT 128  // D

static inline unsigned cdiv(long a, long b) { return (unsigned)((a + b - 1) / b); }

// ---------------------------------------------------------------------------
// deg/dis: dis starts zeroed (memset); accumulate edge counts; finalize adds
// the self-loop (+1) and takes rsqrt.  deg is computed from `row` only, per
// the reference's symmetric-norm recompute.
// ---------------------------------------------------------------------------
__global__ __launch_bounds__(256) void deg_accum(const int* __restrict__ row,
                                                 float* __restrict__ deg, int E) {
  int e = blockIdx.x * 256 + threadIdx.x;
  if (e < E) atomicAdd(&deg[row[e]], 1.0f);
}

__global__ __launch_bounds__(256) void deg_finalize(float* __restrict__ dis, int n) {
  int i = blockIdx.x * 256 + threadIdx.x;
  if (i < n) dis[i] = rsqrtf(dis[i] + 1.0f);  // +1 for the self-loop
}

// ---------------------------------------------------------------------------
// GEMM  O[n x 128] = H[n x 128] @ W[128 x 128]  using V_WMMA_F32_16X16X4_F32.
//
// Block = 256 threads = 8 waves; each wave owns 32 rows (two A fragments),
// so each B fragment loaded from LDS feeds TWO WMMAs.  Per K-step per wave:
// 2 global b64 A-loads, 8 LDS b64 B-loads, 16 WMMAs.  32 K-steps.
//
// W is staged into LDS PRE-SWIZZLED into the B-fragment layout so that each
// fragment (b.x,b.y) = (W[ka][n], W[ka+1][n]) is a single contiguous
// ds_load_b64 (no register shuffles):
//     Wl2[(r>>1)*256 + n*2 + (r&1)] = W[r][n]
// fragment address (dwords) = khalf*256 + l16*2 + k*64 + t*32
//
// A frag (16x4 f32, 2 VGPR): lane l<16 -> A[M=l][k],A[M=l][k+1];
//                            lane l+16 -> A[M=l][k+2],A[M=l][k+3].
// C/D (16x16 f32, 8 VGPR):   acc[t][r] = O[row0 + r + 8*(lane>=16)][t*16+(lane&15)]
// ---------------------------------------------------------------------------
__global__ __launch_bounds__(256) void gemm_wmma_f32(const float* __restrict__ H,
                                                     const float* __restrict__ W,
                                                     float* __restrict__ O,
                                                     int nrows) {
  __shared__ float Wl2[FEAT * FEAT];  // 64 KB, swizzled to fragment layout
  const int tid = threadIdx.x;
  for (int i = tid; i < FEAT * FEAT; i += 256) {
    const int r = i >> 7, n = i & 127;
    Wl2[((r >> 1) << 8) + (n << 1) + (r & 1)] = W[i];  // coalesced global read
  }
  __syncthreads();

  const int wave  = tid >> 5;
  const int lane  = tid & 31;
  const int l16   = lane & 15;
  const int khalf = lane >> 4;  // 0: K=k..k+1, 1: K=k+2..k+3
  const long row0 = (long)blockIdx.x * 256 + wave * 32;

  // Clamp read rows for tail tiles (writes guarded); keeps EXEC all-1s across
  // the WMMA loop as the ISA requires.
  long r0 = row0 + l16, r1 = row0 + 16 + l16;
  const int a0r = (r0 < nrows) ? (int)r0 : (nrows - 1);
  const int a1r = (r1 < nrows) ? (int)r1 : (nrows - 1);
  const float* __restrict__ h0 = H + (size_t)a0r * FEAT;
  const float* __restrict__ h1 = H + (size_t)a1r * FEAT;

  const float* __restrict__ wb = Wl2 + (khalf << 8) + (l16 << 1);

  v8f acc0[8] = {};
  v8f acc1[8] = {};

  for (int k = 0; k < FEAT; k += 4) {
    const int ka = k + khalf * 2;
    const v2f a0 = *(const v2f*)(h0 + ka);  // 8B-aligned (ka even)
    const v2f a1 = *(const v2f*)(h1 + ka);
    const float* wk = wb + (k << 6);
#pragma unroll
    for (int t = 0; t < 8; ++t) {
      const v2f b = *(const v2f*)(wk + (t << 5));  // single ds_load_b64
      acc0[t] = __builtin_amdgcn_wmma_f32_16x16x4_f32(
          false, a0, false, b, (short)0, acc0[t], false, false);
      acc1[t] = __builtin_amdgcn_wmma_f32_16x16x4_f32(
          false, a1, false, b, (short)0, acc1[t], false, false);
    }
  }

#pragma unroll
  for (int t = 0; t < 8; ++t) {
#pragma unroll
    for (int r = 0; r < 8; ++r) {
      const long m0 = row0 + r + khalf * 8;
      if (m0 < nrows) O[(size_t)m0 * FEAT + t * 16 + l16] = acc0[t][r];
      const long m1 = row0 + 16 + r + khalf * 8;
      if (m1 < nrows) O[(size_t)m1 * FEAT + t * 16 + l16] = acc1[t][r];
    }
  }
}

// ---------------------------------------------------------------------------
// Propagate: for each edge e in [0, E+N): out[row] += dis[row]*dis[col]*h[col].
// e >= E are the appended self-loops (row = col = e - E).  One wave per edge:
// 32 lanes x float4 = 128 features, coalesced b128 gather + f32 atomic scatter
// (output array fits in L2, so atomics resolve at L2 rate).
// ---------------------------------------------------------------------------
__global__ __launch_bounds__(256) void scatter_edges(const int* __restrict__ row,
                                                     const int* __restrict__ col,
                                                     const float* __restrict__ dis,
                                                     const float* __restrict__ Hin,
                                                     float* __restrict__ Out,
                                                     int Etot, int Ereal) {
  int e = blockIdx.x * 8 + (threadIdx.x >> 5);
  if (e >= Etot) return;
  const int lane = threadIdx.x & 31;
  int r, c;
  if (e < Ereal) { r = row[e]; c = col[e]; }
  else           { r = c = e - Ereal; }
  const float w = dis[r] * dis[c];
  const float4 v = ((const float4*)(Hin + (size_t)c * FEAT))[lane];
  float* o = Out + (size_t)r * FEAT + lane * 4;
  atomicAdd(o + 0, v.x * w);
  atomicAdd(o + 1, v.y * w);
  atomicAdd(o + 2, v.z * w);
  atomicAdd(o + 3, v.w * w);
}

__global__ __launch_bounds__(256) void bias_relu(float* __restrict__ H,
                                                 const float* __restrict__ bias,
                                                 long total) {
  long i = (long)blockIdx.x * 256 + threadIdx.x;
  if (i < total) {
    float v = H[i] + bias[i & (FEAT - 1)];
    H[i] = v > 0.0f ? v : 0.0f;
  }
}

// ---------------------------------------------------------------------------
// Global mean pool: scatter-add node features into d_out[batch], count nodes,
// divide.  Sum buffer (1000x128 = 512KB) lives in L2.
// ---------------------------------------------------------------------------
__global__ __launch_bounds__(256) void pool_scatter(const float* __restrict__ H,
                                                    const int* __restrict__ batch,
                                                    float* __restrict__ Osum,
                                                    float* __restrict__ cnt, int n) {
  int node = blockIdx.x * 8 + (threadIdx.x >> 5);
  if (node >= n) return;
  const int lane = threadIdx.x & 31;
  const int b = batch[node];
  const float4 v = ((const float4*)(H + (size_t)node * FEAT))[lane];
  float* o = Osum + (size_t)b * FEAT + lane * 4;
  atomicAdd(o + 0, v.x);
  atomicAdd(o + 1, v.y);
  atomicAdd(o + 2, v.z);
  atomicAdd(o + 3, v.w);
  if (lane == 0) atomicAdd(&cnt[b], 1.0f);
}

__global__ __launch_bounds__(256) void pool_div(float* __restrict__ O,
                                                const float* __restrict__ cnt,
                                                int total) {
  int i = blockIdx.x * 256 + threadIdx.x;
  if (i < total) O[i] /= fmaxf(cnt[i >> 7], 1.0f);
}

// ---------------------------------------------------------------------------
extern "C" void kernel_launch(void* const* d_in, const int* in_sizes, int n_in,
                              void* d_out, int out_size, void* d_ws, size_t ws_size,
                              hipStream_t stream) {
  (void)n_in; (void)ws_size;

  const float* x     = (const float*)d_in[0];
  const int*   ei    = (const int*)d_in[1];   // [2, E] row-major: row then col
  const int*   batch = (const int*)d_in[2];
  const float* Wm[3] = {(const float*)d_in[3], (const float*)d_in[5], (const float*)d_in[7]};
  const float* bv[3] = {(const float*)d_in[4], (const float*)d_in[6], (const float*)d_in[8]};

  const int  N  = in_sizes[0] / FEAT;     // 50000
  const int  E  = in_sizes[1] / 2;        // 800000
  const int  Bg = out_size / FEAT;        // 1000
  const int* row = ei;
  const int* col = ei + E;

  // Workspace layout (floats): bufA [N*D] | bufB [N*D] | dis [N] | cnt [B]
  float* bufA = (float*)d_ws;
  float* bufB = bufA + (size_t)N * FEAT;
  float* dis  = bufB + (size_t)N * FEAT;
  float* cnt  = dis + N;

  // --- degree / normalization (row indices are layer-invariant) ---
  hipMemsetAsync(dis, 0, (size_t)N * sizeof(float), stream);
  deg_accum<<<cdiv(E, 256), 256, 0, stream>>>(row, dis, E);
  deg_finalize<<<cdiv(N, 256), 256, 0, stream>>>(dis, N);

  const int Etot = E + N;
  for (int l = 0; l < 3; ++l) {
    const float* hin = (l == 0) ? x : bufB;
    gemm_wmma_f32<<<cdiv(N, 256), 256, 0, stream>>>(hin, Wm[l], bufA, N);
    hipMemsetAsync(bufB, 0, (size_t)N * FEAT * sizeof(float), stream);
    scatter_edges<<<cdiv(Etot, 8), 256, 0, stream>>>(row, col, dis, bufA, bufB, Etot, E);
    bias_relu<<<cdiv((long)N * FEAT, 256), 256, 0, stream>>>(bufB, bv[l], (long)N * FEAT);
  }

  // --- global mean pool ---
  hipMemsetAsync(d_out, 0, (size_t)out_size * sizeof(float), stream);
  hipMemsetAsync(cnt, 0, (size_t)Bg * sizeof(float), stream);
  pool_scatter<<<cdiv(N, 8), 256, 0, stream>>>(bufB, batch, (float*)d_out, cnt, N);
  pool_div<<<cdiv(out_size, 256), 256, 0, stream>>>((float*)d_out, cnt, out_size);
}